// InductiveGraphSAGE_47880295415963
// MI455X (gfx1250) — compile-verified
//
#include <hip/hip_runtime.h>
#include <math.h>

// ---------------------------------------------------------------------------
// InductiveGraphSAGE on MI455X (gfx1250, wave32).
//  * project-then-aggregate (linear commutes with segment-mean): halves edge
//    gather/atomic traffic; P/S accumulators are L2-resident (192 MB L2).
//  * f32 WMMA (V_WMMA_F32_16X16X4_F32) GEMMs:
//      - weight strip staged in LDS, double buffered; async-to-LDS DMA
//        (GLOBAL_LOAD_ASYNC_TO_LDS_B128) when the toolchain exposes it,
//        reg->LDS staging otherwise
//      - A fragments preloaded per chunk in one load clause (wait overlap)
//      - B fragments rotated one k-step ahead from LDS (branchless, unrolled)
//  * fused epilogues (bias + mean + relu), wave-shuffle L2 row normalize.
// ---------------------------------------------------------------------------

typedef __attribute__((ext_vector_type(2))) float v2f;
typedef __attribute__((ext_vector_type(8))) float v8f;

#define FIN 512
#define WPB 8                  // waves per block (256 threads)
#define KT 64                  // K-chunk staged in LDS
#define NSTEP (KT / 4)         // 16 WMMA k-steps per chunk
#define LDSW (KT + 4)          // 68 floats: rows 16B-aligned; b64 frag reads
                               // hit banks {4*mrow+kl ..+1}: all 64 banks once

#if defined(__HIP_DEVICE_COMPILE__)
#if __has_builtin(__builtin_amdgcn_global_load_async_to_lds_b128)
#define USE_ASYNC_LDS 1
#endif
#endif

#if defined(USE_ASYNC_LDS)
// Exact pointee type from the compiler diagnostic: int __vector_size__(16)
typedef int v4i_t __attribute__((__vector_size__(4 * sizeof(int))));
typedef __attribute__((address_space(1))) v4i_t* g_v4i_ptr;   // global (AS1)
typedef __attribute__((address_space(3))) v4i_t* l_v4i_ptr;   // LDS (AS3)

__device__ __forceinline__ void async_cp16(const float* gsrc, const float* ldst) {
    // Global: full 64-bit VA. LDS generic VA: low 32 bits are the LDS offset.
    g_v4i_ptr gp = (g_v4i_ptr)(uintptr_t)gsrc;
    l_v4i_ptr lp = (l_v4i_ptr)(uintptr_t)(unsigned)(uintptr_t)ldst;
    __builtin_amdgcn_global_load_async_to_lds_b128(gp, lp, 0, 0);
}
__device__ __forceinline__ void wait_async0() {
#if __has_builtin(__builtin_amdgcn_s_wait_asynccnt)
    __builtin_amdgcn_s_wait_asynccnt(0);
#else
    asm volatile("s_wait_asynccnt 0x0" ::: "memory");
#endif
}
#endif

__global__ __launch_bounds__(256) void fill_zero_kernel(float* __restrict__ p, int n) {
    int i = blockIdx.x * blockDim.x + threadIdx.x;
    if (i < n) p[i] = 0.0f;
}

// C[M,N] = A[M,K] @ W[N,K]^T (+ epilogue).  K must be a multiple of KT.
//   mode 0: C = A W^T
//   mode 1: C = relu(A W^T + S/max(deg,1) + bias)
//   mode 2: C = A W^T + S/max(deg,1) + bias
// Block: 8 waves = 128 rows x 64 cols; one shared 64xK weight strip via LDS.
__global__ __launch_bounds__(256) void gemm_wmma_f32_kernel(
    const float* __restrict__ A, const float* __restrict__ W,
    float* __restrict__ C,
    const float* __restrict__ S, const float* __restrict__ deg,
    const float* __restrict__ bias,
    int M, int N, int K, int mode)
{
    __shared__ float Wlds[2][64 * LDSW];

    const int tid  = threadIdx.x;
    const int wid  = tid >> 5;
    const int lane = tid & 31;
    const int half = lane >> 4;          // 0: K=k+0,k+1   1: K=k+2,k+3
    const int mrow = lane & 15;

    const int n0   = blockIdx.y * 64;
    const int row0 = (blockIdx.x * WPB + wid) * 16;
    const int rowA = min(row0 + mrow, M - 1);     // clamp OOB waves (stores guarded)

    // W staging map: 4 x 16B per thread per chunk; 16 float4 per 64-float row
    int wr[4], wc[4];
    #pragma unroll
    for (int j = 0; j < 4; ++j) {
        const int idx = tid + j * 256;
        wr[j] = idx >> 4;
        wc[j] = (idx & 15) << 2;
    }

    // ---- prologue: stage chunk 0 into buffer 0 ----
#if defined(USE_ASYNC_LDS)
    #pragma unroll
    for (int j = 0; j < 4; ++j)
        async_cp16(W + (size_t)(n0 + wr[j]) * K + wc[j],
                   &Wlds[0][wr[j] * LDSW + wc[j]]);
    wait_async0();
#else
    float4 wreg[4];
    #pragma unroll
    for (int j = 0; j < 4; ++j)
        wreg[j] = *(const float4*)(W + (size_t)(n0 + wr[j]) * K + wc[j]);
    #pragma unroll
    for (int j = 0; j < 4; ++j)
        *(float4*)(&Wlds[0][wr[j] * LDSW + wc[j]]) = wreg[j];
#endif
    __syncthreads();

    v8f acc0 = {}, acc1 = {}, acc2 = {}, acc3 = {};

    const float* Ab = A + (size_t)rowA * K + 2 * half;
    const int nChunks = K / KT;

    for (int c = 0; c < nChunks; ++c) {
        const int buf  = c & 1;
        const bool more = (c + 1 < nChunks);

        // issue next chunk's weight staging; completion hidden under 64 WMMAs
        if (more) {
            const int kcn = (c + 1) * KT;
#if defined(USE_ASYNC_LDS)
            #pragma unroll
            for (int j = 0; j < 4; ++j)
                async_cp16(W + (size_t)(n0 + wr[j]) * K + kcn + wc[j],
                           &Wlds[buf ^ 1][wr[j] * LDSW + wc[j]]);
#else
            #pragma unroll
            for (int j = 0; j < 4; ++j)
                wreg[j] = *(const float4*)(W + (size_t)(n0 + wr[j]) * K + kcn + wc[j]);
#endif
            __builtin_prefetch(Ab + (size_t)(c + 1) * KT, 0, 1);  // global_prefetch_b8
        }

        // preload the whole A chunk as one clause of 16 b64 loads (in-order
        // completion => partial waits overlap with early WMMAs)
        v2f afrag[NSTEP];
        #pragma unroll
        for (int i = 0; i < NSTEP; ++i)
            afrag[i] = *(const v2f*)(Ab + c * KT + i * 4);

        const float* lds = &Wlds[buf][0];

        // preload B fragments for k-step 0, then rotate one step ahead
        v2f bc[4], bn[4];
        #pragma unroll
        for (int t = 0; t < 4; ++t)
            bc[t] = *(const v2f*)(lds + (t * 16 + mrow) * LDSW + 2 * half);

        #pragma unroll
        for (int step = 0; step < NSTEP; ++step) {
            if (step + 1 < NSTEP) {
                const int kl = (step + 1) * 4 + 2 * half;
                #pragma unroll
                for (int t = 0; t < 4; ++t)
                    bn[t] = *(const v2f*)(lds + (t * 16 + mrow) * LDSW + kl);
            }
            acc0 = __builtin_amdgcn_wmma_f32_16x16x4_f32(false, afrag[step], false, bc[0], (short)0, acc0, false, false);
            acc1 = __builtin_amdgcn_wmma_f32_16x16x4_f32(false, afrag[step], false, bc[1], (short)0, acc1, false, false);
            acc2 = __builtin_amdgcn_wmma_f32_16x16x4_f32(false, afrag[step], false, bc[2], (short)0, acc2, false, false);
            acc3 = __builtin_amdgcn_wmma_f32_16x16x4_f32(false, afrag[step], false, bc[3], (short)0, acc3, false, false);
            if (step + 1 < NSTEP) {
                #pragma unroll
                for (int t = 0; t < 4; ++t) bc[t] = bn[t];
            }
        }

#if defined(USE_ASYNC_LDS)
        if (more) { wait_async0(); }
        __syncthreads();
#else
        __syncthreads();
        if (more) {
            #pragma unroll
            for (int j = 0; j < 4; ++j)
                *(float4*)(&Wlds[buf ^ 1][wr[j] * LDSW + wc[j]]) = wreg[j];
        }
        __syncthreads();
#endif
    }

    if (row0 >= M) return;   // wave-uniform

    v8f accs[4] = {acc0, acc1, acc2, acc3};
    // C/D layout: VGPR j -> row j (lanes 0-15) / row j+8 (lanes 16-31), col = lane&15
    #pragma unroll
    for (int j = 0; j < 8; ++j) {
        const int m = row0 + j + 8 * half;
        float invd = 1.0f;
        if (mode != 0) invd = 1.0f / fmaxf(deg[m], 1.0f);
        #pragma unroll
        for (int t = 0; t < 4; ++t) {
            const int col = n0 + t * 16 + mrow;
            float v = accs[t][j];
            if (mode != 0) v += S[(size_t)m * N + col] * invd + bias[col];
            if (mode == 1) v = fmaxf(v, 0.0f);
            C[(size_t)m * N + col] = v;
        }
    }
}

// One wave per edge: S[dst] += P[src] (F floats, float4 gathers), optional degree.
__global__ __launch_bounds__(256) void edge_scatter_kernel(
    const float* __restrict__ P, const int* __restrict__ ei,
    float* __restrict__ S, float* __restrict__ deg,
    int E, int F)
{
    const int wid  = threadIdx.x >> 5;
    const int lane = threadIdx.x & 31;
    const int e = blockIdx.x * WPB + wid;
    if (e >= E) return;
    const int src = ei[e];
    const int dst = ei[E + e];
    const float* p = P + (size_t)src * F;
    float*       s = S + (size_t)dst * F;
    for (int f = lane * 4; f < F; f += 128) {
        const float4 v = *(const float4*)(p + f);
        atomicAdd(&s[f + 0], v.x);
        atomicAdd(&s[f + 1], v.y);
        atomicAdd(&s[f + 2], v.z);
        atomicAdd(&s[f + 3], v.w);
    }
    if (deg != nullptr && lane == 0)
        atomicAdd(&deg[dst], 1.0f);
}

// One wave per row: out = T / max(||T||_2, 1e-12)
__global__ __launch_bounds__(256) void l2norm_rows_kernel(
    const float* __restrict__ T, float* __restrict__ out, int M, int F)
{
    const int wid  = threadIdx.x >> 5;
    const int lane = threadIdx.x & 31;
    const int r = blockIdx.x * WPB + wid;
    if (r >= M) return;
    const float* t = T + (size_t)r * F;
    float v[4];
    float ss = 0.0f;
    const int cnt = F >> 5;   // F=128 -> 4
    #pragma unroll
    for (int i = 0; i < 4; ++i)
        if (i < cnt) { v[i] = t[lane + 32 * i]; ss += v[i] * v[i]; }
    #pragma unroll
    for (int off = 16; off > 0; off >>= 1)
        ss += __shfl_xor(ss, off, 32);
    const float inv = 1.0f / fmaxf(sqrtf(ss), 1e-12f);
    #pragma unroll
    for (int i = 0; i < 4; ++i)
        if (i < cnt) out[(size_t)r * F + lane + 32 * i] = v[i] * inv;
}

static inline int ceil_div(long long a, long long b) { return (int)((a + b - 1) / b); }

extern "C" void kernel_launch(void* const* d_in, const int* in_sizes, int n_in,
                              void* d_out, int out_size, void* d_ws, size_t ws_size,
                              hipStream_t stream) {
    const float* x   = (const float*)d_in[0];
    const int*   ei  = (const int*)  d_in[1];
    const float* W1l = (const float*)d_in[2];
    const float* b1  = (const float*)d_in[3];
    const float* W1r = (const float*)d_in[4];
    const float* W2l = (const float*)d_in[5];
    const float* b2  = (const float*)d_in[6];
    const float* W2r = (const float*)d_in[7];

    const int M  = in_sizes[0] / FIN;   // 50000
    const int E  = in_sizes[1] / 2;     // 800000
    const int FH = in_sizes[3];         // 256
    const int FO = in_sizes[6];         // 128

    float* ws  = (float*)d_ws;
    float* S1  = ws;                              // M*FH
    float* P1  = S1 + (size_t)M * FH;             // M*FH (later reused as T)
    float* H   = P1 + (size_t)M * FH;             // M*FH
    float* deg = H  + (size_t)M * FH;             // M
    float* P2  = S1;                              // M*FO (reuses S1 after layer 1)
    float* S2  = S1 + (size_t)M * FO;             // M*FO
    float* T   = P1;                              // M*FO (reuses P1)
    float* outp = (float*)d_out;

    const dim3 blk(256);
    const int rowBlocks = ceil_div(M, 16 * WPB);  // 128 rows per block

    // ---- layer 1 ----
    fill_zero_kernel<<<ceil_div((long long)M * FH, 256), blk, 0, stream>>>(S1, M * FH);
    fill_zero_kernel<<<ceil_div(M, 256), blk, 0, stream>>>(deg, M);

    // P1 = x @ W1_l^T
    gemm_wmma_f32_kernel<<<dim3(rowBlocks, FH / 64), blk, 0, stream>>>(
        x, W1l, P1, nullptr, nullptr, nullptr, M, FH, FIN, 0);
    edge_scatter_kernel<<<ceil_div(E, WPB), blk, 0, stream>>>(P1, ei, S1, deg, E, FH);
    // H = relu(x @ W1_r^T + S1/deg + b1)
    gemm_wmma_f32_kernel<<<dim3(rowBlocks, FH / 64), blk, 0, stream>>>(
        x, W1r, H, S1, deg, b1, M, FH, FIN, 1);

    // ---- layer 2 ----
    fill_zero_kernel<<<ceil_div((long long)M * FO, 256), blk, 0, stream>>>(S2, M * FO);
    // P2 = H @ W2_l^T
    gemm_wmma_f32_kernel<<<dim3(rowBlocks, FO / 64), blk, 0, stream>>>(
        H, W2l, P2, nullptr, nullptr, nullptr, M, FO, FH, 0);
    edge_scatter_kernel<<<ceil_div(E, WPB), blk, 0, stream>>>(P2, ei, S2, nullptr, E, FO);
    // T = H @ W2_r^T + S2/deg + b2
    gemm_wmma_f32_kernel<<<dim3(rowBlocks, FO / 64), blk, 0, stream>>>(
        H, W2r, T, S2, deg, b2, M, FO, FH, 2);

    // ---- row-wise L2 normalize ----
    l2norm_rows_kernel<<<ceil_div(M, WPB), blk, 0, stream>>>(T, outp, M, FO);
}